// SNN_19224273617385
// MI455X (gfx1250) — compile-verified
//
#include <hip/hip_runtime.h>
#include <hip/hip_bf16.h>

#define B_   256
#define T_   1024
#define DIN  32
#define DH   200
#define DOUT 2
#define ROWS (B_ * T_)          // 262144 flattened (b,t) rows
#define ZPITCH 208              // padded LDS row pitch for 200 cols

typedef float v2f __attribute__((ext_vector_type(2)));
typedef float v8f __attribute__((ext_vector_type(8)));

// decay constants: expf(-1/10), expf(-1/20) computed in f32 like the reference
#define A1_DECAY 0.9048374180359595f
#define A2_DECAY 0.9512294245007140f

// ---------------------------------------------------------------------------
// Fused layer-1 kernel: one block per batch element b.
// 416 threads = 13 waves; wave w owns the 16-column N-tile w of D_h=200.
// For each chunk of 16 consecutive timesteps:
//   1) stage X[b, t0:t0+16, :] (16x32 f32) into LDS cooperatively
//   2) each wave: 8x v_wmma_f32_16x16x4_f32 -> z1 tile (bias folded into C)
//   3) stage z1 tile to LDS
//   4) threads 0..199 (h = tid) run the 16-step membrane recurrence from LDS,
//      writing s1 and m1 directly to their final output locations.
// W1 B-fragments + bias are loop-invariant per wave: preloaded into registers
// and reused across all 64 chunks.
// ---------------------------------------------------------------------------
__global__ __launch_bounds__(416) void snn_fused1(const float* __restrict__ X,
                                                  const float* __restrict__ W1,
                                                  const float* __restrict__ b1,
                                                  float* __restrict__ S1,
                                                  float* __restrict__ M1) {
    __shared__ float x_lds[16 * DIN];      // 2 KB: 16 timesteps x 32 inputs
    __shared__ float z_lds[16 * ZPITCH];   // 13.3 KB: 16 timesteps x 200 (+pad)

    const int b    = blockIdx.x;
    const int tid  = threadIdx.x;
    const int lane = tid & 31;
    const int w    = tid >> 5;             // wave id == N-tile id (0..12)
    const int lm   = lane & 15;            // M row (A) / N col (B,C,D)
    const int kh   = (lane >> 4) * 2;      // K sub-offset 0 / 2
    const int col  = w * 16 + lm;
    const bool cv  = (col < DH);
    const int colL = cv ? col : (DH - 1);

    // Loop-invariant B-fragments (W1 columns) + bias, resident in VGPRs.
    v2f bf[8];
#pragma unroll
    for (int kk = 0; kk < DIN / 4; ++kk) {
        const int kb = kk * 4 + kh;
        v2f t;
        t.x = W1[(size_t)kb * DH + colL];
        t.y = W1[(size_t)(kb + 1) * DH + colL];
        if (!cv) { t.x = 0.0f; t.y = 0.0f; }
        bf[kk] = t;
    }
    float binit = b1[colL];
    if (!cv) binit = 0.0f;

    const int h = tid;                     // scan role: hidden neuron index
    float mh = 0.0f;                       // membrane state, carried in-register

    const size_t xbase = (size_t)b * T_ * DIN;
    const size_t obase = (size_t)b * T_ * DH;

    for (int chunk = 0; chunk < T_ / 16; ++chunk) {
        const int t0 = chunk * 16;

        // ---- 1) cooperative X tile load (512 floats, 128 lanes x float4)
        if (tid < 128) {
            const float4 v =
                *(const float4*)(X + xbase + (size_t)t0 * DIN + tid * 4);
            *(float4*)&x_lds[tid * 4] = v;
        }
        __syncthreads();

        // ---- 2) WMMA: z1 tile = Xchunk @ W1 + b1
        v8f c;
#pragma unroll
        for (int v = 0; v < 8; ++v) c[v] = binit;
#pragma unroll
        for (int kk = 0; kk < DIN / 4; ++kk) {
            const int kb = kk * 4 + kh;                 // even -> 8B aligned
            const v2f a = *(const v2f*)&x_lds[lm * DIN + kb];
            c = __builtin_amdgcn_wmma_f32_16x16x4_f32(false, a, false, bf[kk],
                                                      (short)0, c, false, false);
        }

        // ---- 3) stage z1 tile to LDS (row = t-in-chunk, col = h)
        if (cv) {
            const int rhalf = (lane >> 4) << 3;         // +8 for hi lanes
#pragma unroll
            for (int v = 0; v < 8; ++v)
                z_lds[(rhalf + v) * ZPITCH + col] = c[v];
        }
        __syncthreads();

        // ---- 4) 16-step membrane recurrence, direct output writes
        if (h < DH) {
            float m = mh;
#pragma unroll
            for (int tt = 0; tt < 16; ++tt) {
                const float z = z_lds[tt * ZPITCH + h];
                m = m * A1_DECAY + z;
                const float s = (m > 0.5f) ? 1.0f : 0.0f;  // spike(m-0.5 > 0)
                m -= 0.5f * s;                             // soft reset
                const size_t gi = obase + (size_t)(t0 + tt) * DH + h;
                S1[gi] = s;
                M1[gi] = m;
            }
            mh = m;
        }
        __syncthreads();   // protect x_lds/z_lds before next chunk
    }
}

// ---------------------------------------------------------------------------
// Kernel 3: U2 = s1 @ W2 + b2   (f32 WMMA, K=200 -> 50 steps, N padded to 16,
// only cols 0..1 stored).  U2 is staged into the s2 output region.
// ---------------------------------------------------------------------------
__global__ __launch_bounds__(128) void snn_gemm2(const float* __restrict__ S1,
                                                 const float* __restrict__ W2,
                                                 const float* __restrict__ b2,
                                                 float* __restrict__ U2) {
    const int lane = threadIdx.x & 31;
    const int wave = blockIdx.x * 4 + (threadIdx.x >> 5);
    const int row0 = wave * 16;
    if (row0 >= ROWS) return;                  // wave-uniform guard

    const int lm = lane & 15;
    const int kh = (lane >> 4) * 2;
    const bool cv = (lm < DOUT);

    float binit = b2[lm & 1];
    if (!cv) binit = 0.0f;
    v8f c;
#pragma unroll
    for (int v = 0; v < 8; ++v) c[v] = binit;

    const float* arow = S1 + (size_t)(row0 + lm) * DH;
#pragma unroll 10
    for (int kk = 0; kk < DH / 4; ++kk) {
        const int kb = kk * 4 + kh;
        v2f a, b;
        a.x = arow[kb];
        a.y = arow[kb + 1];
        b.x = W2[(size_t)kb * DOUT + (lm & 1)];
        b.y = W2[(size_t)(kb + 1) * DOUT + (lm & 1)];
        if (!cv) { b.x = 0.0f; b.y = 0.0f; }
        c = __builtin_amdgcn_wmma_f32_16x16x4_f32(false, a, false, b,
                                                  (short)0, c, false, false);
    }

    if (cv) {
        const int rbase = row0 + ((lane >> 4) << 3);
#pragma unroll
        for (int v = 0; v < 8; ++v)
            U2[(size_t)(rbase + v) * DOUT + lm] = c[v];
    }
}

// ---------------------------------------------------------------------------
// Kernel 4: layer-2 scan + output accumulation. One thread per (b,o).
// us holds U2 on input, overwritten with s2 spikes.
// output = sum_{t>=1} m2[t] / T
// ---------------------------------------------------------------------------
__global__ __launch_bounds__(256) void snn_scan2(float* us, float* m2out,
                                                 float* out) {
    const int tid = blockIdx.x * blockDim.x + threadIdx.x;
    if (tid >= B_ * DOUT) return;
    const int b = tid / DOUT;
    const int o = tid % DOUT;
    const size_t base = (size_t)b * T_ * DOUT + o;

    float m = 0.0f, acc = 0.0f;
    for (int t = 0; t < T_; ++t) {
        const size_t idx = base + (size_t)t * DOUT;
        const float u = us[idx];
        m = m * A2_DECAY + u;
        const float s = (m > 1.0f) ? 1.0f : 0.0f;
        m -= s;                                     // soft reset (thresh=1)
        us[idx]    = s;
        m2out[idx] = m;
        if (t > 0) acc += m;
    }
    out[tid] = acc * (1.0f / (float)T_);
}

// ---------------------------------------------------------------------------
extern "C" void kernel_launch(void* const* d_in, const int* in_sizes, int n_in,
                              void* d_out, int out_size, void* d_ws, size_t ws_size,
                              hipStream_t stream) {
    const float* x  = (const float*)d_in[0];   // [256,1024,32]
    const float* W1 = (const float*)d_in[1];   // [32,200]
    const float* b1 = (const float*)d_in[2];   // [200]
    const float* W2 = (const float*)d_in[3];   // [200,2]
    const float* b2 = (const float*)d_in[4];   // [2]

    float* out = (float*)d_out;                          // [256,2]
    float* s1  = out + (size_t)B_ * DOUT;                // [256,1024,200]
    float* m1  = s1 + (size_t)B_ * T_ * DH;              // [256,1024,200]
    float* s2  = m1 + (size_t)B_ * T_ * DH;              // [256,1024,2]
    float* m2  = s2 + (size_t)B_ * T_ * DOUT;            // [256,1024,2]

    // 1) fused: s1, m1 = scan(X@W1 + b1)  -- one block per batch element
    snn_fused1<<<B_, 416, 0, stream>>>(x, W1, b1, s1, m1);

    // 2) U2 = s1@W2 + b2, staged into the s2 region (same shape/layout)
    const int waves2 = ROWS / 16;                        // 16384
    snn_gemm2<<<waves2 / 4, 128, 0, stream>>>(s1, W2, b2, s2);

    // 3) layer-2 scan: reads U2 from s2 region, writes s2, m2, output
    snn_scan2<<<(B_ * DOUT + 255) / 256, 256, 0, stream>>>(s2, m2, out);
}